// RPFQModule_2602750181485
// MI455X (gfx1250) — compile-verified
//
#include <hip/hip_runtime.h>
#include <hip/hip_bf16.h>

typedef __attribute__((ext_vector_type(2))) float v2f;
typedef __attribute__((ext_vector_type(8))) float v8f;

#define PLANE      65536          // 256*256 pixels per channel plane
#define S_COEF     (10000.0f / 280.0f)   // S_SCALE / OUT_DIV
#define T_COEF     (300.0f   / 280.0f)   // T_SCALE / OUT_DIV
#define OFF_SCALE  4.0f
#define Q_SCALE    0.05f

__device__ __forceinline__ float bperm(float v, int srcLane) {
    // Dst[lane] = src[srcLane]; byte-addressed index. Must be called with
    // full EXEC (kept as unconditional straight-line statements).
    return __int_as_float(__builtin_amdgcn_ds_bpermute(srcLane << 2, __float_as_int(v)));
}

__global__ __launch_bounds__(256) void posmap_affine_wmma(
    const float* __restrict__ Off,   // (B,3,256,256)
    const float* __restrict__ Q,     // (B,4)
    const float* __restrict__ T,     // (B,3)
    const float* __restrict__ mean,  // (3,256,256)
    float* __restrict__ out,         // (B,3,256,256)
    int wavesPerBatch, int itersPerWave)
{
    const int tid   = threadIdx.x;
    const int lane  = tid & 31;
    const int half  = lane >> 4;     // 0: lanes 0-15, 1: lanes 16-31
    const int m     = lane & 15;

    const int gwave       = blockIdx.x * (blockDim.x >> 5) + (tid >> 5);
    const int batch       = gwave / wavesPerBatch;
    const int waveInBatch = gwave % wavesPerBatch;

    // ---- per-batch 3x4 transform: coeff[d][k<3] = rot[d][k]*(k==1?-1:1)*S/280,
    //      coeff[d][3] = T[d]*300/280 (uniform per wave -> scalarized) ----
    const float q0 = Q[batch*4+0]*Q_SCALE, q1 = Q[batch*4+1]*Q_SCALE,
                q2 = Q[batch*4+2]*Q_SCALE, q3 = Q[batch*4+3]*Q_SCALE;
    const float ww = q0*q0, xx = q1*q1, yy = q2*q2, zz = q3*q3;
    const float c00 =  (ww + xx - yy - zz)     * S_COEF;
    const float c01 = -(2.0f*(q1*q2 + q0*q3))  * S_COEF;
    const float c02 =  (2.0f*(q1*q3 - q0*q2))  * S_COEF;
    const float c10 =  (2.0f*(q1*q2 - q0*q3))  * S_COEF;
    const float c11 = -(ww - xx + yy - zz)     * S_COEF;
    const float c12 =  (2.0f*(q0*q1 + q2*q3))  * S_COEF;
    const float c20 =  (2.0f*(q0*q2 + q1*q3))  * S_COEF;
    const float c21 = -(2.0f*(q2*q3 - q0*q1))  * S_COEF;
    const float c22 =  (ww - xx - yy + zz)     * S_COEF;
    const float c03 = T[batch*3+0] * T_COEF;
    const float c13 = T[batch*3+1] * T_COEF;
    const float c23 = T[batch*3+2] * T_COEF;

    // ---- A matrix (16x4 f32): lane-half 0 holds K=0,1; half 1 holds K=2,3.
    //      Rows M=0..2 = output channels; rows 3..15 zero. ----
    float a0 = (m == 0) ? (half ? c02 : c00)
             : (m == 1) ? (half ? c12 : c10)
             : (m == 2) ? (half ? c22 : c20) : 0.0f;
    float a1 = (m == 0) ? (half ? c03 : c01)
             : (m == 1) ? (half ? c13 : c11)
             : (m == 2) ? (half ? c23 : c21) : 0.0f;
    const v2f A = { a0, a1 };

    const size_t base = (size_t)batch * 3 * PLANE;
    const float* ox = Off + base;
    const float* oy = ox + PLANE;
    const float* oz = oy + PLANE;
    const float* mx = mean;
    const float* my = mean + PLANE;
    const float* mz = mean + 2*PLANE;
    float* o0 = out + base;
    float* o1 = o0 + PLANE;
    float* o2 = o1 + PLANE;

    const int p0   = waveInBatch * itersPerWave * 32;
    const bool lo  = (lane < 16);

    for (int i = 0; i < itersPerWave; ++i) {
        const int p = p0 + i * 32 + lane;   // 32 consecutive pixels per wave-iter

        // streamed reads (NT) + L2-resident mean (RT): 3x128B fully coalesced
        float x = __builtin_nontemporal_load(ox + p) * OFF_SCALE + mx[p];
        float y = __builtin_nontemporal_load(oy + p) * OFF_SCALE + my[p];
        float z = __builtin_nontemporal_load(oz + p) * OFF_SCALE + mz[p];

        if (i + 8 < itersPerWave) {
            // near-scope prefetch 8 iters (1KB/plane) ahead; addresses in-bounds
            __builtin_prefetch(ox + (p + 8*32), 0, 3);
            __builtin_prefetch(oy + (p + 8*32), 0, 3);
            __builtin_prefetch(oz + (p + 8*32), 0, 3);
        }

        // Repack 32 pixels into two B tiles (4x16 each):
        //   V0 = K0 row (x, lanes 0-15) | K2 row (z, lanes 16-31)
        //   V1 = K1 row (y)             | K3 row (1.0)
        // bpermutes are unconditional (full EXEC); selection via cndmask only.
        float zlo = bperm(z, m);          // upper half needs z of pixels 0-15
        float xhi = bperm(x, lane | 16);  // lower half needs x of pixels 16-31
        float yhi = bperm(y, lane | 16);
        const v2f Ba = { lo ? x   : zlo, lo ? y   : 1.0f };
        const v2f Bb = { lo ? xhi : z,   lo ? yhi : 1.0f };

        v8f acc = {};
        // D = A(3x4 transform) x B(4x16 homogeneous pixels); rows 0..2 = channels
        v8f dA = __builtin_amdgcn_wmma_f32_16x16x4_f32(
            false, A, false, Ba, (short)0, acc, false, false);
        v8f dB = __builtin_amdgcn_wmma_f32_16x16x4_f32(
            false, A, false, Bb, (short)0, acc, false, false);

        // Merge: lanes 0-15 <- tile A pixels, lanes 16-31 <- tile B pixels.
        // Hoist bpermutes out of the selects so no exec-mask branching is born.
        float sb0 = bperm(dB[0], m);
        float sb1 = bperm(dB[1], m);
        float sb2 = bperm(dB[2], m);
        float r0 = lo ? dA[0] : sb0;
        float r1 = lo ? dA[1] : sb1;
        float r2 = lo ? dA[2] : sb2;

        // 3x128B coalesced NT stores to planar output
        __builtin_nontemporal_store(r0, o0 + p);
        __builtin_nontemporal_store(r1, o1 + p);
        __builtin_nontemporal_store(r2, o2 + p);
    }
}

extern "C" void kernel_launch(void* const* d_in, const int* in_sizes, int n_in,
                              void* d_out, int out_size, void* d_ws, size_t ws_size,
                              hipStream_t stream) {
    const float* Off  = (const float*)d_in[0];   // (B,3,256,256)
    const float* Q    = (const float*)d_in[1];   // (B,4)
    const float* T    = (const float*)d_in[2];   // (B,3)
    const float* mean = (const float*)d_in[3];   // (3,256,256)
    float* out = (float*)d_out;

    const int B = in_sizes[1] / 4;
    const int pixels = in_sizes[3] / 3;          // 65536
    const int wavesPerBatch = 64;
    const int itersPerWave  = pixels / (32 * wavesPerBatch);  // 32

    const int wavesPerBlock = 8;                 // 256 threads, wave32
    dim3 grid((B * wavesPerBatch) / wavesPerBlock);
    dim3 block(wavesPerBlock * 32);

    posmap_affine_wmma<<<grid, block, 0, stream>>>(
        Off, Q, T, mean, out, wavesPerBatch, itersPerWave);
}